// GAT_20521353741094
// MI455X (gfx1250) — compile-verified
//
#include <hip/hip_runtime.h>
#include <math.h>

typedef __attribute__((ext_vector_type(2))) float v2f;
typedef __attribute__((ext_vector_type(8))) float v8f;

#define NEG_SLOPE 0.2f

// ---------------------------------------------------------------------------
// Tiny matvec: v[k] = sum_j W[k*128+j] * a[j]   (W is [128,128], a is [128])
// ---------------------------------------------------------------------------
__global__ void matvec128(const float* __restrict__ W, const float* __restrict__ a,
                          float* __restrict__ v) {
  int k = threadIdx.x; // 128 threads
  const float4* wr = (const float4*)(W + (size_t)k * 128);
  const float4* av = (const float4*)a;
  float acc = 0.f;
#pragma unroll
  for (int i = 0; i < 32; ++i) {
    float4 w = wr[i], x = av[i];
    acc += w.x * x.x + w.y * x.y + w.z * x.z + w.w * x.w;
  }
  v[k] = acc;
}

// v2[k] = W[k*2+0]*a[0] + W[k*2+1]*a[1]   (W is [128,2])
__global__ void matvec128x2(const float* __restrict__ W, const float* __restrict__ a,
                            float* __restrict__ v) {
  int k = threadIdx.x; // 128 threads
  v[k] = W[k * 2 + 0] * a[0] + W[k * 2 + 1] * a[1];
}

// ---------------------------------------------------------------------------
// Fused dual GEMM (fp32 WMMA):  xs = x @ Wsrc ;  h = x @ Wl + (bl + b1)
// x:[N,128], weights:[128,128]. One block = 16 rows; 8 waves = 8 col strips.
// Uses V_WMMA_F32_16X16X4_F32; K-loop of 32 steps, two accumulators share A.
// ---------------------------------------------------------------------------
__global__ __launch_bounds__(256) void gemm_l1(
    const float* __restrict__ x, const float* __restrict__ Wsrc,
    const float* __restrict__ Wl, const float* __restrict__ bl,
    const float* __restrict__ b1, float* __restrict__ xs, float* __restrict__ h) {
  const int wave = threadIdx.x >> 5;  // 0..7 : column tile
  const int lane = threadIdx.x & 31;
  const int l15  = lane & 15;
  const int kg   = lane >> 4;         // lane group 0/1 -> K sub-pair
  const int row0 = blockIdx.x * 16;
  const int col0 = wave * 16;
  const int col  = col0 + l15;

  const float* Arow = x + (size_t)(row0 + l15) * 128;

  v8f acc0 = {};
  v8f acc1 = {};
#pragma unroll
  for (int k = 0; k < 128; k += 4) {
    const int ka = k + kg * 2;
    v2f a;  a.x  = Arow[ka];                     a.y  = Arow[ka + 1];
    v2f b0; b0.x = Wsrc[(size_t)ka * 128 + col]; b0.y = Wsrc[(size_t)(ka + 1) * 128 + col];
    v2f b1v; b1v.x = Wl[(size_t)ka * 128 + col]; b1v.y = Wl[(size_t)(ka + 1) * 128 + col];
    acc0 = __builtin_amdgcn_wmma_f32_16x16x4_f32(false, a, false, b0, (short)0, acc0, false, false);
    acc1 = __builtin_amdgcn_wmma_f32_16x16x4_f32(false, a, false, b1v, (short)0, acc1, false, false);
  }

  // C/D layout: VGPR r, lanes 0-15 -> row row0+r ; lanes 16-31 -> row row0+8+r
  const int rbase = row0 + kg * 8;
  const float bias = bl[col] + b1[col];
#pragma unroll
  for (int r = 0; r < 8; ++r) {
    xs[(size_t)(rbase + r) * 128 + col] = acc0[r];
    h [(size_t)(rbase + r) * 128 + col] = acc1[r] + bias;
  }
}

// ---------------------------------------------------------------------------
// Node prep (layer 1): ssrc = xs.a_src ; sdst = x.vdst ; m=-inf ; s=0
// ---------------------------------------------------------------------------
__global__ void node_prep1(const float* __restrict__ x, const float* __restrict__ xs,
                           const float* __restrict__ a_src, const float* __restrict__ vdst,
                           float* __restrict__ ssrc, float* __restrict__ sdst,
                           float* __restrict__ m, float* __restrict__ s, int N) {
  int n = blockIdx.x * blockDim.x + threadIdx.x;
  if (n >= N) return;
  const float4* xv  = (const float4*)(x  + (size_t)n * 128);
  const float4* xsv = (const float4*)(xs + (size_t)n * 128);
  const float4* av  = (const float4*)a_src;
  const float4* vv  = (const float4*)vdst;
  float d1 = 0.f, d2 = 0.f;
#pragma unroll 4
  for (int i = 0; i < 32; ++i) {
    float4 a = xsv[i], b = av[i];
    d1 += a.x * b.x + a.y * b.y + a.z * b.z + a.w * b.w;
    float4 c = xv[i], d = vv[i];
    d2 += c.x * d.x + c.y * d.y + c.z * d.z + c.w * d.w;
  }
  ssrc[n] = d1; sdst[n] = d2;
  m[n] = -__builtin_inff();
  s[n] = 0.f;
}

// ---------------------------------------------------------------------------
// Edge pass A: logit + leaky relu, atomic segment max over dst
// ---------------------------------------------------------------------------
__global__ void edge_logit(const int* __restrict__ src, const int* __restrict__ dst,
                           const float* __restrict__ ssrc, const float* __restrict__ sdst,
                           float* __restrict__ ebuf, float* __restrict__ m, int E) {
  int i = blockIdx.x * blockDim.x + threadIdx.x;
  if (i >= E) return;
  int d = dst[i];
  float e = ssrc[src[i]] + sdst[d];
  e = e > 0.f ? e : NEG_SLOPE * e;
  ebuf[i] = e;
  __hip_atomic_fetch_max(&m[d], e, __ATOMIC_RELAXED, __HIP_MEMORY_SCOPE_AGENT);
}

// ---------------------------------------------------------------------------
// Edge pass B: p = exp(e - m[dst]); atomic sum over dst; ebuf <- p
// ---------------------------------------------------------------------------
__global__ void edge_softmax(const int* __restrict__ dst, float* __restrict__ ebuf,
                             const float* __restrict__ m, float* __restrict__ s, int E) {
  int i = blockIdx.x * blockDim.x + threadIdx.x;
  if (i >= E) return;
  int d = dst[i];
  float p = __expf(ebuf[i] - m[d]);
  ebuf[i] = p;
  __hip_atomic_fetch_add(&s[d], p, __ATOMIC_RELAXED, __HIP_MEMORY_SCOPE_AGENT);
}

// ---------------------------------------------------------------------------
// Edge aggregation, 128 channels: one wave per edge, float4 per lane
// ---------------------------------------------------------------------------
__global__ __launch_bounds__(256) void edge_agg128(
    const int* __restrict__ src, const int* __restrict__ dst,
    const float* __restrict__ p, const float* __restrict__ s,
    const float* __restrict__ xs, float* __restrict__ out, int E) {
  int e = blockIdx.x * 8 + (threadIdx.x >> 5);
  if (e >= E) return;
  int lane = threadIdx.x & 31;
  int sn = src[e], dn = dst[e];
  float alpha = p[e] / (s[dn] + 1e-16f);
  float4 v = ((const float4*)(xs + (size_t)sn * 128))[lane];
  float* o = out + (size_t)dn * 128 + lane * 4;
  __hip_atomic_fetch_add(o + 0, alpha * v.x, __ATOMIC_RELAXED, __HIP_MEMORY_SCOPE_AGENT);
  __hip_atomic_fetch_add(o + 1, alpha * v.y, __ATOMIC_RELAXED, __HIP_MEMORY_SCOPE_AGENT);
  __hip_atomic_fetch_add(o + 2, alpha * v.z, __ATOMIC_RELAXED, __HIP_MEMORY_SCOPE_AGENT);
  __hip_atomic_fetch_add(o + 3, alpha * v.w, __ATOMIC_RELAXED, __HIP_MEMORY_SCOPE_AGENT);
}

// ---------------------------------------------------------------------------
// Elementwise ReLU on N*128 floats (float4 vectorized)
// ---------------------------------------------------------------------------
__global__ void relu_vec(float4* __restrict__ h, int n4) {
  int i = blockIdx.x * blockDim.x + threadIdx.x;
  if (i >= n4) return;
  float4 v = h[i];
  v.x = fmaxf(v.x, 0.f); v.y = fmaxf(v.y, 0.f);
  v.z = fmaxf(v.z, 0.f); v.w = fmaxf(v.w, 0.f);
  h[i] = v;
}

// ---------------------------------------------------------------------------
// Layer-2 node kernel: hs2 = h@W2_src ; out = h@Wl2 + bl2 + b2 ;
// ssrc2 = hs2 . a2_src ; sdst2 = h . vdst2 ; m=-inf ; s=0
// ---------------------------------------------------------------------------
__global__ void node_prep2(const float* __restrict__ h, const float* __restrict__ W2src,
                           const float* __restrict__ Wl2, const float* __restrict__ bl2,
                           const float* __restrict__ b2, const float* __restrict__ a2src,
                           const float* __restrict__ vdst2, float* __restrict__ hs2,
                           float* __restrict__ out, float* __restrict__ ssrc,
                           float* __restrict__ sdst, float* __restrict__ m,
                           float* __restrict__ s, int N) {
  int n = blockIdx.x * blockDim.x + threadIdx.x;
  if (n >= N) return;
  const float4* hv = (const float4*)(h + (size_t)n * 128);
  const float2* w2 = (const float2*)W2src;
  const float2* wl = (const float2*)Wl2;
  const float4* vv = (const float4*)vdst2;
  float s0 = 0.f, s1 = 0.f, t0 = 0.f, t1 = 0.f, dd = 0.f;
#pragma unroll 4
  for (int i = 0; i < 32; ++i) {
    float4 a = hv[i];
    float4 v = vv[i];
    int k = i * 4;
    float2 wa = w2[k], wb = w2[k + 1], wc = w2[k + 2], wd = w2[k + 3];
    s0 += a.x * wa.x + a.y * wb.x + a.z * wc.x + a.w * wd.x;
    s1 += a.x * wa.y + a.y * wb.y + a.z * wc.y + a.w * wd.y;
    float2 la = wl[k], lb = wl[k + 1], lc = wl[k + 2], ld = wl[k + 3];
    t0 += a.x * la.x + a.y * lb.x + a.z * lc.x + a.w * ld.x;
    t1 += a.x * la.y + a.y * lb.y + a.z * lc.y + a.w * ld.y;
    dd += a.x * v.x + a.y * v.y + a.z * v.z + a.w * v.w;
  }
  hs2[n * 2 + 0] = s0;
  hs2[n * 2 + 1] = s1;
  out[n * 2 + 0] = t0 + bl2[0] + b2[0];
  out[n * 2 + 1] = t1 + bl2[1] + b2[1];
  ssrc[n] = s0 * a2src[0] + s1 * a2src[1];
  sdst[n] = dd;
  m[n] = -__builtin_inff();
  s[n] = 0.f;
}

// ---------------------------------------------------------------------------
// Edge aggregation, 2 channels: one thread per edge
// ---------------------------------------------------------------------------
__global__ void edge_agg2(const int* __restrict__ src, const int* __restrict__ dst,
                          const float* __restrict__ p, const float* __restrict__ s,
                          const float* __restrict__ hs2, float* __restrict__ out, int E) {
  int i = blockIdx.x * blockDim.x + threadIdx.x;
  if (i >= E) return;
  int sn = src[i], dn = dst[i];
  float alpha = p[i] / (s[dn] + 1e-16f);
  float2 v = ((const float2*)hs2)[sn];
  __hip_atomic_fetch_add(&out[dn * 2 + 0], alpha * v.x, __ATOMIC_RELAXED, __HIP_MEMORY_SCOPE_AGENT);
  __hip_atomic_fetch_add(&out[dn * 2 + 1], alpha * v.y, __ATOMIC_RELAXED, __HIP_MEMORY_SCOPE_AGENT);
}

// ---------------------------------------------------------------------------
// Final activation: sigmoid(relu(x)) in place
// ---------------------------------------------------------------------------
__global__ void final_act(float* __restrict__ out, int n) {
  int i = blockIdx.x * blockDim.x + threadIdx.x;
  if (i >= n) return;
  float v = fmaxf(out[i], 0.f);
  out[i] = 1.f / (1.f + __expf(-v));
}

// ---------------------------------------------------------------------------
extern "C" void kernel_launch(void* const* d_in, const int* in_sizes, int n_in,
                              void* d_out, int out_size, void* d_ws, size_t ws_size,
                              hipStream_t stream) {
  const float* x      = (const float*)d_in[0];
  const int*   eidx   = (const int*)d_in[1];
  const float* W1_src = (const float*)d_in[2];
  const float* W1_dst = (const float*)d_in[3];
  const float* a1_src = (const float*)d_in[4];
  const float* a1_dst = (const float*)d_in[5];
  const float* b1     = (const float*)d_in[6];
  const float* Wl1    = (const float*)d_in[7];
  const float* bl1    = (const float*)d_in[8];
  const float* W2_src = (const float*)d_in[9];
  const float* W2_dst = (const float*)d_in[10];
  const float* a2_src = (const float*)d_in[11];
  const float* a2_dst = (const float*)d_in[12];
  const float* b2     = (const float*)d_in[13];
  const float* Wl2    = (const float*)d_in[14];
  const float* bl2    = (const float*)d_in[15];

  const int N = in_sizes[0] / 128;
  const int E = in_sizes[1] / 2;
  const int* src = eidx;
  const int* dst = eidx + E;

  float* out = (float*)d_out;

  // workspace layout (floats)
  float* ws = (float*)d_ws;
  size_t off = 0;
  float* xs1   = ws + off; off += (size_t)N * 128; // layer1 xs / reused region
  float* h     = ws + off; off += (size_t)N * 128;
  float* ebuf  = ws + off; off += (size_t)E;
  float* ssrc  = ws + off; off += (size_t)N;
  float* sdst  = ws + off; off += (size_t)N;
  float* mbuf  = ws + off; off += (size_t)N;
  float* sbuf  = ws + off; off += (size_t)N;
  float* hs2   = ws + off; off += (size_t)N * 2;
  float* vdst1 = ws + off; off += 128;
  float* vdst2 = ws + off; off += 128;

  const int TB = 256;
  const int nblkN = (N + TB - 1) / TB;
  const int nblkE = (E + TB - 1) / TB;

  // ---- Layer 1 ----
  matvec128<<<1, 128, 0, stream>>>(W1_dst, a1_dst, vdst1);
  gemm_l1<<<N / 16, 256, 0, stream>>>(x, W1_src, Wl1, bl1, b1, xs1, h);
  node_prep1<<<nblkN, TB, 0, stream>>>(x, xs1, a1_src, vdst1, ssrc, sdst, mbuf, sbuf, N);
  edge_logit<<<nblkE, TB, 0, stream>>>(src, dst, ssrc, sdst, ebuf, mbuf, E);
  edge_softmax<<<nblkE, TB, 0, stream>>>(dst, ebuf, mbuf, sbuf, E);
  edge_agg128<<<(E + 7) / 8, 256, 0, stream>>>(src, dst, ebuf, sbuf, xs1, h, E);
  relu_vec<<<((N * 32) + TB - 1) / TB, TB, 0, stream>>>((float4*)h, N * 32);

  // ---- Layer 2 ----
  matvec128x2<<<1, 128, 0, stream>>>(W2_dst, a2_dst, vdst2);
  node_prep2<<<nblkN, TB, 0, stream>>>(h, W2_src, Wl2, bl2, b2, a2_src, vdst2,
                                       hs2, out, ssrc, sdst, mbuf, sbuf, N);
  edge_logit<<<nblkE, TB, 0, stream>>>(src, dst, ssrc, sdst, ebuf, mbuf, E);
  edge_softmax<<<nblkE, TB, 0, stream>>>(dst, ebuf, mbuf, sbuf, E);
  edge_agg2<<<nblkE, TB, 0, stream>>>(src, dst, ebuf, sbuf, hs2, out, E);
  final_act<<<((N * 2) + TB - 1) / TB, TB, 0, stream>>>(out, N * 2);
}